// Quantizer_49297634623863
// MI455X (gfx1250) — compile-verified
//
#include <hip/hip_runtime.h>
#include <hip/hip_bf16.h>

typedef __attribute__((ext_vector_type(2))) float v2f;
typedef __attribute__((ext_vector_type(8))) float v8f;
typedef __attribute__((ext_vector_type(4))) unsigned int u32x4;
typedef __attribute__((ext_vector_type(8))) int i32x8;
typedef __attribute__((ext_vector_type(4))) int i32x4;

#define K_CODES     1024
#define D_DIM       256
#define HW          1024     // 32*32 per image
#define ROWS_PER_WG 64
#define SLICE       32       // weight columns per TDM slice
#define N_KG        (K_CODES / SLICE)
#define ZSTR        65       // z tile LDS row stride   (64 + 1 pad, via TDM pad)
#define WSTR        33       // w slice LDS row stride  (32 + 1 pad, via TDM pad)

// ---------------------------------------------------------------------------
// Kernel 1: wsq[k] = sum_d weight[d][k]^2   (coalesced over k)
// ---------------------------------------------------------------------------
__global__ void vq_wsq_kernel(const float* __restrict__ w, float* __restrict__ wsq) {
  int k = blockIdx.x * blockDim.x + threadIdx.x;
  if (k < K_CODES) {
    float s = 0.f;
#pragma unroll 8
    for (int d = 0; d < D_DIM; ++d) {
      float v = w[(size_t)d * K_CODES + k];
      s += v * v;
    }
    wsq[k] = s;
  }
}

// ---------------------------------------------------------------------------
// TDM descriptor helpers (CDNA5 ISA ch.8: D# groups; clang-23 6-arg builtin).
// Tensor: [256 rows (d)] x [1024 cols], dim0 contiguous, stride 1024 elems.
// Tile:   [TILE0 cols] x [256 d]; LDS gets [d][TILE0 + 1 pad] f32 rows.
// ---------------------------------------------------------------------------
// group1: wg_mask=0 | data_size=2(4B) | pad_enable | pad_interval | pad_amount=0(1 DW)
//         tensor_dim0=1024, tensor_dim1=256, tile_dim0, tile_dim1=256,
//         tensor_dim0_stride=1024, tensor_dim1_stride=0
#define TDM_G1(PAD_IV, TILE0)                                            \
  { (int)((2u << 16) | (1u << 20) | ((unsigned)(PAD_IV) << 22)),         \
    (int)(1024u << 16), (int)(256u << 16), (int)((unsigned)(TILE0) << 16),\
    256, 1024, 0, 0 }

__device__ __forceinline__ void tdm_load_tile(unsigned int lds_byte_addr,
                                              const float* gptr,
                                              i32x8 g1) {
  unsigned long long ga = (unsigned long long)(uintptr_t)gptr;
  u32x4 g0;
  g0[0] = 1u;                                                   // count=1 (valid D#)
  g0[1] = lds_byte_addr;                                        // lds_addr
  g0[2] = (unsigned int)ga;                                     // global_addr[31:0]
  g0[3] = (unsigned int)((ga >> 32) & 0x01FFFFFFu) | (2u << 30);// ga[56:32] | type=2
  i32x4 gz4 = {0, 0, 0, 0};
  i32x8 gz8 = {0, 0, 0, 0, 0, 0, 0, 0};
  __builtin_amdgcn_tensor_load_to_lds(g0, g1, gz4, gz4, gz8, 0); // TENSORcnt-tracked
}

// ---------------------------------------------------------------------------
// Kernel 2: per 64-pixel block: TDM-staged fp32 WMMA scores, argmin, gather,
//           mse, losses. 128 threads = 4 waves; wave wv owns rows wv*16..+15.
// ---------------------------------------------------------------------------
__launch_bounds__(128)
__global__ void vq_main_kernel(const float* __restrict__ z,
                               const float* __restrict__ weight,
                               const float* __restrict__ wsq,
                               float* __restrict__ out_q,
                               float* __restrict__ out_loss,
                               float* __restrict__ out_commit,
                               float* __restrict__ out_embed) {
  __shared__ float lds_zt[D_DIM * ZSTR];       // z tile  [d][row], TDM-padded
  __shared__ float lds_wt[2][D_DIM * WSTR];    // w slice [d][col], double-buffered
  __shared__ float lds_wsq[K_CODES];
  __shared__ int   lds_kbest[ROWS_PER_WG];
  __shared__ float lds_part[128];

  const int tid   = threadIdx.x;
  const int nbase = blockIdx.x * ROWS_PER_WG;   // first flat pixel index
  const int b     = nbase >> 10;                // 64 | 1024 -> one image per block
  const int hw0   = nbase & (HW - 1);
  const size_t zbase = (size_t)b * (D_DIM * HW) + hw0;

  const int lane = tid & 31;
  const int wv   = tid >> 5;
  const int m    = lane & 15;   // row slot (A) / col slot (B)
  const int hi   = lane >> 4;   // half-wave -> K-pair {0,1} vs {2,3}
  const int m0   = wv * 16;

  const i32x8 g1z = TDM_G1(5, 64);    // 64-DWORD rows + 1 pad -> stride 65
  const i32x8 g1w = TDM_G1(4, SLICE); // 32-DWORD rows + 1 pad -> stride 33

  // Kick off async DMA: z tile + first weight slice (wave 0 issues; EXEC ignored)
  if (wv == 0) {
    tdm_load_tile((unsigned int)(uintptr_t)&lds_zt[0],    z + zbase, g1z);
    tdm_load_tile((unsigned int)(uintptr_t)&lds_wt[0][0], weight,    g1w);
  }
  // stage wsq with plain loads, overlapped with the DMA
  for (int i = tid; i < K_CODES; i += 128) lds_wsq[i] = wsq[i];

  float best_s[8];
  int   best_k[8];
#pragma unroll
  for (int j = 0; j < 8; ++j) { best_s[j] = 3.4e38f; best_k[j] = 0; }

#define SCORE_UPDATE(CREG, COL)                                         \
  {                                                                     \
    const int colg = (COL);                                             \
    const float wq = lds_wsq[colg];                                     \
    _Pragma("unroll")                                                   \
    for (int j = 0; j < 8; ++j) {                                       \
      float sc = wq - 2.0f * (CREG)[j];                                 \
      if (sc < best_s[j]) { best_s[j] = sc; best_k[j] = colg; }         \
    }                                                                   \
  }

  const float* zr = &lds_zt[m0 + m];   // A row pointer: + dd*ZSTR picks d

  for (int kg = 0; kg < N_KG; ++kg) {
    if (wv == 0) __builtin_amdgcn_s_wait_tensorcnt(0);  // tile for kg is in LDS
    __syncthreads();
    // prefetch next slice into the other buffer (last read 2 barriers ago)
    if (wv == 0 && kg + 1 < N_KG)
      tdm_load_tile((unsigned int)(uintptr_t)&lds_wt[(kg + 1) & 1][0],
                    weight + (kg + 1) * SLICE, g1w);

    const float* wb = lds_wt[kg & 1];
    v8f c0 = {}, c1 = {};
#pragma unroll 4
    for (int s = 0; s < D_DIM / 4; ++s) {
      const int dd = 4 * s + 2 * hi;
      // A 16x4 f32 frag: lane(m,hi) holds z[d=dd][row m0+m], z[dd+1][...]
      v2f a  = { zr[dd * ZSTR],           zr[(dd + 1) * ZSTR] };
      // B 4x16 f32 frags: lane(m,hi) holds w[dd][col], w[dd+1][col]
      v2f b0 = { wb[dd * WSTR + m],       wb[(dd + 1) * WSTR + m] };
      v2f b1 = { wb[dd * WSTR + 16 + m],  wb[(dd + 1) * WSTR + 16 + m] };
      c0 = __builtin_amdgcn_wmma_f32_16x16x4_f32(false, a, false, b0, (short)0, c0, false, false);
      c1 = __builtin_amdgcn_wmma_f32_16x16x4_f32(false, a, false, b1, (short)0, c1, false, false);
    }

    const int cb = kg * SLICE + m;
    SCORE_UPDATE(c0, cb)
    SCORE_UPDATE(c1, cb + 16)
  }
#undef SCORE_UPDATE

  // cross-lane argmin within each 16-lane half (cols live across lanes);
  // tie -> lower k, matching jnp.argmin first-min semantics
#pragma unroll
  for (int j = 0; j < 8; ++j) {
#pragma unroll
    for (int off = 1; off < 16; off <<= 1) {
      float os = __shfl_xor(best_s[j], off, 32);
      int   ok = __shfl_xor(best_k[j], off, 32);
      if (os < best_s[j] || (os == best_s[j] && ok < best_k[j])) {
        best_s[j] = os; best_k[j] = ok;
      }
    }
  }
  if (m == 0) {
    // C layout: VGPR j holds row m0+j (lanes 0-15) / row m0+8+j (lanes 16-31)
#pragma unroll
    for (int j = 0; j < 8; ++j) lds_kbest[m0 + hi * 8 + j] = best_k[j];
  }
  __syncthreads();

  // Epilogue: gather winning code column (L2-hot), write quantized coalesced
  // over hw, accumulate per-row squared error (row fixed: 128 % 64 == 0).
  const int r  = tid & 63;
  const int kb = lds_kbest[r];
  float acc = 0.f;
  for (int i = tid; i < ROWS_PER_WG * D_DIM; i += 128) {
    int d = i >> 6;
    float wvv = weight[(size_t)d * K_CODES + kb];
    float zv  = lds_zt[d * ZSTR + r];
    float df  = wvv - zv;
    acc += df * df;
    out_q[zbase + (size_t)d * HW + r] = wvv;   // q_st forward == quantized
  }
  lds_part[tid] = acc;
  __syncthreads();
  if (tid < ROWS_PER_WG) {
    float mse = (lds_part[tid] + lds_part[tid + 64]) * (1.0f / D_DIM);
    int n = nbase + tid;
    out_loss[n]   = 1.25f * mse;   // beta*commit + embed, commit == embed numerically
    out_commit[n] = mse;
    out_embed[n]  = mse;
  }
}

// ---------------------------------------------------------------------------
extern "C" void kernel_launch(void* const* d_in, const int* in_sizes, int n_in,
                              void* d_out, int out_size, void* d_ws, size_t ws_size,
                              hipStream_t stream) {
  const float* z      = (const float*)d_in[0];   // [32,256,32,32] f32
  const float* weight = (const float*)d_in[1];   // [256,1024] f32
  float* wsq = (float*)d_ws;                     // 1024 floats scratch

  float* out        = (float*)d_out;
  float* out_q      = out;                           // 8388608
  float* out_loss   = out + 8388608;                 // 32768
  float* out_commit = out + 8388608 + 32768;         // 32768
  float* out_embed  = out + 8388608 + 65536;         // 32768

  vq_wsq_kernel<<<K_CODES / 128, 128, 0, stream>>>(weight, wsq);
  vq_main_kernel<<<(32 * 1024) / ROWS_PER_WG, 128, 0, stream>>>(
      z, weight, wsq, out_q, out_loss, out_commit, out_embed);
}